// FeatureFunMax_80255758893451
// MI455X (gfx1250) — compile-verified
//
#include <hip/hip_runtime.h>
#include <hip/hip_bf16.h>

// Problem constants (reference: B=8, T=2048, H=256, L=16)
#define BB 8
#define TT 2048
#define HH 256
#define LL 16
#define TILE_T 64                 // t-rows per phase-2 block
#define HALO_LO 15                // l goes back up to 15 rows
#define HALO_HI 16                // relay looks 16 rows ahead
#define ROWS (TILE_T + HALO_LO + HALO_HI)   // 95 staged rows
#define LSTRIDE (HH + 4)          // 260: pads rows so l-lanes hit distinct LDS banks

typedef float v2f __attribute__((ext_vector_type(2)));
typedef float v8f __attribute__((ext_vector_type(8)));
typedef int   v4i __attribute__((ext_vector_type(4)));

#if defined(__HIP_DEVICE_COMPILE__)
#if !__has_builtin(__builtin_amdgcn_wmma_f32_16x16x4_f32)
#error "device pass: missing __builtin_amdgcn_wmma_f32_16x16x4_f32"
#endif
#if __has_builtin(__builtin_amdgcn_global_load_async_to_lds_b128)
#define HAVE_ASYNC_LDS 1
#endif
#if __has_builtin(__builtin_amdgcn_s_wait_asynccnt)
#define HAVE_WAIT_ASYNC_BUILTIN 1
#endif
#endif

#if defined(HAVE_ASYNC_LDS)
typedef __attribute__((address_space(1))) v4i* gp_v4i;   // global (AS1), non-const
typedef __attribute__((address_space(3))) v4i* lp_v4i;   // LDS (AS3)
#endif

__device__ __forceinline__ void wait_async_zero() {
#if defined(__HIP_DEVICE_COMPILE__)
#if defined(HAVE_WAIT_ASYNC_BUILTIN)
    __builtin_amdgcn_s_wait_asynccnt(0);
#elif defined(HAVE_ASYNC_LDS)
    asm volatile("s_wait_asynccnt 0" ::: "memory");
#endif
#endif
}

// ---------------------------------------------------------------------------
// Phase 1: G[r,l]  = sum_h logits[r,h] * whole_w[l,h]        (true GEMM)
//          G2[r,0] = sum_h logits[r,h] * end_w[h]
//          G2[r,1] = sum_h logits[r,h] * relay_w[h]
// One wave per 16-row tile of r; V_WMMA_F32_16X16X4_F32 over K=256 (64 steps).
// ---------------------------------------------------------------------------
__global__ __launch_bounds__(256) void phase1_wmma(
    const float* __restrict__ logits,   // [B*T, 256]
    const float* __restrict__ whole_w,  // [16, 256]
    const float* __restrict__ end_w,    // [256]
    const float* __restrict__ relay_w,  // [256]
    float* __restrict__ G,              // [B*T, 16]
    float* __restrict__ G2)             // [B*T, 16] (cols 0,1 used)
{
#if defined(__HIP_DEVICE_COMPILE__)
    const int lane  = threadIdx.x & 31;
    const int wave  = threadIdx.x >> 5;                 // 0..7
    const int tile  = blockIdx.x * 8 + wave;            // 1024 tiles
    const int row0  = tile * 16;

    const int m  = lane & 15;          // A: M index / B: N index
    const int kh = (lane >> 4) << 1;   // 0 or 2 (K sub-offset)

    const float* Arow = logits  + (size_t)(row0 + m) * HH + kh;
    const float* Brow = whole_w + (size_t)m * HH + kh;            // whole_w[n][.]
    // Lanes m>=2 read relay_w (valid memory) and zero the value with a select
    // instead of a divergent (exec-masked) load.
    const float* B2p  = ((m == 0) ? end_w : relay_w) + kh;        // col0/col1
    const bool   b2on = (m < 2);

    v8f accG  = {0.f,0.f,0.f,0.f,0.f,0.f,0.f,0.f};
    v8f accG2 = {0.f,0.f,0.f,0.f,0.f,0.f,0.f,0.f};

#pragma unroll 4
    for (int k0 = 0; k0 < HH; k0 += 4) {
        v2f a   = *(const v2f*)(Arow + k0);
        v2f bw  = *(const v2f*)(Brow + k0);
        v2f b2v = *(const v2f*)(B2p + k0);        // unconditional load
        v2f b2;
        b2.x = b2on ? b2v.x : 0.f;                // v_cndmask, no exec dance
        b2.y = b2on ? b2v.y : 0.f;
        // 8 args: (neg_a, A, neg_b, B, c_mod, C, reuse_a, reuse_b)
        accG  = __builtin_amdgcn_wmma_f32_16x16x4_f32(false, a,  false, bw, (short)0, accG,  false, false);
        accG2 = __builtin_amdgcn_wmma_f32_16x16x4_f32(false, a,  false, b2, (short)0, accG2, false, false);
    }

    // C/D layout: VGPR v -> M = v + 8*(lane>=16), N = lane&15
    const int mbase = (lane >> 4) << 3;
    const int n     = lane & 15;
#pragma unroll
    for (int v = 0; v < 8; ++v) {
        const size_t r = (size_t)(row0 + mbase + v);
        G [r * LL + n] = accG[v];
        G2[r * LL + n] = accG2[v];
    }
#endif
}

// ---------------------------------------------------------------------------
// Phase 2: out[r,l]  = 0.5*(G[r,l] + G[r-l,l] + sum_h w[l,h]*|c-p|) + G2[r,0] + bias[l]
//          relay[r]  = 0.5*(G2[r,1] + G2[r+16,1] + sum_h rw[h]*|c-q|) + relay_b
// Block = 256 threads, covers 64 t-rows; logits tile (+halos) staged in LDS
// via GLOBAL_LOAD_ASYNC_TO_LDS_B128 (ASYNCcnt-tracked).
// thread -> (t-group, l); c-row loads broadcast across the 16 l-lanes.
// ---------------------------------------------------------------------------
__global__ __launch_bounds__(256) void phase2_main(
    const float* __restrict__ logits,
    const float* __restrict__ whole_w,
    const float* __restrict__ relay_w,
    const float* __restrict__ relay_b,
    const float* __restrict__ length_bias,
    const float* __restrict__ G,
    const float* __restrict__ G2,
    float* __restrict__ out,        // [B*T, 16]
    float* __restrict__ relay_out)  // [B*T]
{
    __shared__ float lds[ROWS * LSTRIDE];

    const int tid = threadIdx.x;
    const int blk = blockIdx.x;                 // 256 blocks = 8 b * 32 chunks
    const int b   = blk >> 5;
    const int t0  = (blk & 31) * TILE_T;

    // ---- stage logits rows [t0-15, t0+79] into LDS (zero-fill OOB) ----
    const float4* src = (const float4*)logits;
    for (int i = tid; i < ROWS * (HH / 4); i += 256) {
        const int row = i >> 6;        // staged row index
        const int c4  = i & 63;        // float4 column
        const int grow = t0 - HALO_LO + row;
        float* dst = &lds[row * LSTRIDE + (c4 << 2)];
        if (grow >= 0 && grow < TT) {
            const float4* gp = &src[((size_t)(b * TT + grow) << 6) + c4];
#if defined(HAVE_ASYNC_LDS)
            __builtin_amdgcn_global_load_async_to_lds_b128(
                (gp_v4i)(float4*)gp, (lp_v4i)dst, 0, 0);
#else
            *(float4*)dst = *gp;
#endif
        } else {
            *(float4*)dst = make_float4(0.f, 0.f, 0.f, 0.f);
        }
    }
    wait_async_zero();
    __syncthreads();

    const int l  = tid & 15;
    const int tg = tid >> 4;                    // 0..15
    const float lbias = length_bias[l];
    const float4* wrow = (const float4*)(whole_w + (l << 8));   // L0-cached

#pragma unroll 1
    for (int pass = 0; pass < TILE_T / 16; ++pass) {
        const int tcur  = t0 + (pass << 4) + tg;
        const int lrow  = (tcur - t0) + HALO_LO;
        const float* crow = &lds[lrow * LSTRIDE];
        const float* prow = &lds[(lrow - l) * LSTRIDE];

        float acc = 0.f;
#pragma unroll 8
        for (int j = 0; j < HH / 4; ++j) {
            const float4 c = *(const float4*)(crow + (j << 2));
            const float4 p = *(const float4*)(prow + (j << 2));
            const float4 w = wrow[j];
            acc += w.x * __builtin_fabsf(c.x - p.x)
                 + w.y * __builtin_fabsf(c.y - p.y)
                 + w.z * __builtin_fabsf(c.z - p.z)
                 + w.w * __builtin_fabsf(c.w - p.w);
        }

        const size_t r  = (size_t)b * TT + tcur;
        const float g   = G[(r << 4) + l];
        const float gp  = (tcur >= l) ? G[((r - (size_t)l) << 4) + l] : 0.f;
        const float ev  = G2[r << 4];                 // end score (col 0)
        out[(r << 4) + l] = 0.5f * (acc + g + gp) + ev + lbias;
    }

    // ---- relay branch: one thread per t (64 threads active) ----
    if (tid < TILE_T) {
        const int t = t0 + tid;
        const float* crow = &lds[(tid + HALO_LO) * LSTRIDE];
        const float* qrow = crow + HALO_HI * LSTRIDE;     // row t+16 (zero-padded)
        const float4* rw4 = (const float4*)relay_w;

        float racc = 0.f;
#pragma unroll 8
        for (int j = 0; j < HH / 4; ++j) {
            const float4 c = *(const float4*)(crow + (j << 2));
            const float4 q = *(const float4*)(qrow + (j << 2));
            const float4 w = rw4[j];
            racc += w.x * __builtin_fabsf(c.x - q.x)
                  + w.y * __builtin_fabsf(c.y - q.y)
                  + w.z * __builtin_fabsf(c.z - q.z)
                  + w.w * __builtin_fabsf(c.w - q.w);
        }
        const size_t r = (size_t)b * TT + t;
        const float rc = G2[(r << 4) + 1];
        const float rq = (t + HALO_HI < TT) ? G2[((r + HALO_HI) << 4) + 1] : 0.f;
        relay_out[r] = 0.5f * (racc + rc + rq) + relay_b[0];
    }
}

// ---------------------------------------------------------------------------
extern "C" void kernel_launch(void* const* d_in, const int* in_sizes, int n_in,
                              void* d_out, int out_size, void* d_ws, size_t ws_size,
                              hipStream_t stream) {
    (void)in_sizes; (void)n_in; (void)out_size; (void)ws_size;

    const float* logits      = (const float*)d_in[0];  // [8,2048,256]
    const float* end_w       = (const float*)d_in[1];  // [256]
    const float* whole_w     = (const float*)d_in[2];  // [16,256]
    const float* relay_w     = (const float*)d_in[3];  // [256]
    const float* relay_b     = (const float*)d_in[4];  // scalar
    const float* length_bias = (const float*)d_in[5];  // [16]

    float* out_p   = (float*)d_out;                    // [8,2048,16]
    float* relay_p = out_p + (size_t)BB * TT * LL;     // [8,2048]

    float* G  = (float*)d_ws;                          // [B*T,16] = 1 MB
    float* G2 = G + (size_t)BB * TT * LL;              // [B*T,16] = 1 MB

    // Phase 1: 1024 tiles of 16 rows, 8 waves (256 thr) per block -> 128 blocks
    phase1_wmma<<<dim3(128), dim3(256), 0, stream>>>(logits, whole_w, end_w, relay_w, G, G2);

    // Phase 2: 8 b * 32 t-chunks = 256 blocks of 256 threads
    phase2_main<<<dim3(256), dim3(256), 0, stream>>>(logits, whole_w, relay_w, relay_b,
                                                     length_bias, G, G2, out_p, relay_p);
}